// RankBoost_26310969655743
// MI455X (gfx1250) — compile-verified
//
#include <hip/hip_runtime.h>
#include <hip/hip_bf16.h>

// Problem constants (match reference)
constexpr int kB = 512;
constexpr int kT = 50;
constexpr int kN = 10000;
constexpr int kD = 128;      // DIM
constexpr int kL = 4;
constexpr int kH = 64;
constexpr int kG = 4 * kD;   // 512 gate width
constexpr int kLH = kL * kH; // 256

typedef float v2f __attribute__((ext_vector_type(2)));
typedef float v8f __attribute__((ext_vector_type(8)));

__device__ __forceinline__ float sigmoidf_(float x) {
    return 1.0f / (1.0f + __expf(-x));
}

// --------------------------------------------------------------------------
// V_WMMA_F32_16X16X4_F32 fragments (wave32):
//  A (16x4):  m = lane&15; VGPR0 holds K = k0 + (lane>>4)*2, VGPR1 holds +1
//  B (4x16):  n = lane&15; VGPR0 holds K = k0 + (lane>>4)*2, VGPR1 holds +1
//  C/D (16x16): VGPR v, lane l -> row = v + (l>=16 ? 8:0), col = l&15
// --------------------------------------------------------------------------

// ========================= LSTM (recurrent) ===============================
// One block = 16 batch rows, 256 threads = 8 waves; each wave owns 4 of the
// 32 gate N-tiles. K is the OUTER loop and stays ROLLED: the scheduler then
// batches the 8 B-frag loads into s_clause groups and drains them with
// staggered s_wait_loadcnt (verified in codegen) — unrolling regressed this
// into load->wait(0)->wmma chains. 4 accumulators (32 VGPRs) live; A-frags
// (LDS) reused across the 4 tiles; B-frags stream from L2-resident wih/whh.
__global__ __launch_bounds__(256)
void lstm_kernel(const int* __restrict__ user_seq, const float* __restrict__ emb,
                 const float* __restrict__ wih, const float* __restrict__ whh,
                 const float* __restrict__ bih, const float* __restrict__ bhh,
                 float* __restrict__ hout)
{
    __shared__ float xs[16][kD];   // gathered x_t tile
    __shared__ float hs[16][kD];   // hidden state
    __shared__ float cs[16][kD];   // cell state
    __shared__ float gs[16][kG];   // gate pre-activations

    const int b0    = blockIdx.x * 16;
    const int tid   = threadIdx.x;
    const int lane  = tid & 31;
    const int wave  = tid >> 5;
    const int mrow  = lane & 15;
    const int khalf = (lane >> 4) * 2;
    const int rbase = (lane >> 4) * 8;
    const int c0    = wave * 64 + mrow;   // gate column of this lane's j=0 tile

    for (int i = tid; i < 16 * kD; i += 256) {
        (&hs[0][0])[i] = 0.0f;
        (&cs[0][0])[i] = 0.0f;
    }
    __syncthreads();

    for (int t = 0; t < kT; ++t) {
        // gather embedding rows for this timestep (float4-coalesced)
        for (int i = tid; i < 16 * (kD / 4); i += 256) {
            int r = i >> 5, k4 = i & 31;
            int idx = user_seq[(b0 + r) * kT + t];
            ((float4*)&xs[r][0])[k4] = ((const float4*)(emb + (size_t)idx * kD))[k4];
        }
        __syncthreads();

        v8f acc[4] = {};
        #pragma unroll 1
        for (int k0 = 0; k0 < kD; k0 += 4) {
            const int kb = k0 + khalf;
            const v2f ax = *(const v2f*)&xs[mrow][kb];
            const v2f ah = *(const v2f*)&hs[mrow][kb];
            #pragma unroll
            for (int j = 0; j < 4; ++j) {
                const size_t brow = (size_t)(c0 + j * 16) * kD + kb;
                v2f bx = *(const v2f*)&wih[brow];
                acc[j] = __builtin_amdgcn_wmma_f32_16x16x4_f32(
                            false, ax, false, bx, (short)0, acc[j], false, false);
                v2f bh = *(const v2f*)&whh[brow];
                acc[j] = __builtin_amdgcn_wmma_f32_16x16x4_f32(
                            false, ah, false, bh, (short)0, acc[j], false, false);
            }
        }
        #pragma unroll
        for (int j = 0; j < 4; ++j) {
            const int ncol = c0 + j * 16;
            const float bias = bih[ncol] + bhh[ncol];
            #pragma unroll
            for (int v = 0; v < 8; ++v)
                gs[rbase + v][ncol] = acc[j][v] + bias;
        }
        __syncthreads();

        // elementwise LSTM cell update (torch gate order i,f,g,o)
        for (int i = tid; i < 16 * kD; i += 256) {
            int r = i >> 7, d = i & (kD - 1);
            float ig = sigmoidf_(gs[r][d]);
            float fg = sigmoidf_(gs[r][kD + d]);
            float gg = tanhf(gs[r][2 * kD + d]);
            float og = sigmoidf_(gs[r][3 * kD + d]);
            float c  = fg * cs[r][d] + ig * gg;
            cs[r][d] = c;
            hs[r][d] = og * tanhf(c);
        }
        __syncthreads();
    }

    for (int i = tid; i < 16 * (kD / 4); i += 256) {
        int r = i >> 5, k4 = i & 31;
        ((float4*)(hout + (size_t)(b0 + r) * kD))[k4] = ((float4*)&hs[r][0])[k4];
    }
}

// ===================== weak-learner projection GEMM =======================
// C[M, 256] = A[M,128] @ W1part; column n -> learner l=n>>6, unit h=n&63;
// W1 is [L, 2D, H]; koff selects seq (0) vs item (128) half. One wave per
// 16x16 tile, 8 tiles per block. K loop kept rolled to bound pressure.
__global__ __launch_bounds__(256)
void wl_gemm_kernel(const float* __restrict__ A, int M,
                    const float* __restrict__ W1, int koff,
                    const float* __restrict__ b1, int addBias,
                    float* __restrict__ C)
{
    const int tid  = threadIdx.x;
    const int lane = tid & 31;
    const int wave = tid >> 5;
    const int tilesN = kLH / 16;             // 16
    const int tile = blockIdx.x * 8 + wave;
    const int mt = tile / tilesN;
    const int nt = tile % tilesN;
    if (mt * 16 >= M) return;

    const int m0 = mt * 16, n0 = nt * 16;
    const int mrow  = lane & 15;
    const int khalf = (lane >> 4) * 2;
    const int ncol  = n0 + mrow;
    const int l  = ncol >> 6;
    const int hh = ncol & 63;
    const float* w1col = W1 + (size_t)(l * (2 * kD) + koff) * kH + hh;

    v8f acc = {};
    #pragma unroll 1
    for (int k0 = 0; k0 < kD; k0 += 4) {
        const int kb = k0 + khalf;
        v2f a = *(const v2f*)&A[(size_t)(m0 + mrow) * kD + kb];
        v2f b;
        b.x = w1col[(size_t)kb * kH];
        b.y = w1col[(size_t)(kb + 1) * kH];
        acc = __builtin_amdgcn_wmma_f32_16x16x4_f32(
                false, a, false, b, (short)0, acc, false, false);
    }
    const float bias = addBias ? b1[ncol] : 0.0f;
    const int rbase = (lane >> 4) * 8;
    #pragma unroll
    for (int v = 0; v < 8; ++v)
        C[(size_t)(m0 + rbase + v) * kLH + ncol] = acc[v] + bias;
}

// ========================= fused relu-dot scores ==========================
// score[b,n] = cadd + sum_k relu(sp[b,k] + ip[n,k]) * (alpha[k>>6]*W2[k])
// VALU-bound by construction (relu breaks GEMM structure). 16x64 tile per
// block; LDS row stride 260 keeps float4 alignment AND rotates banks by 4
// per row (4*16 = 64 -> conflict-free for the 16-row broadcast pattern).
constexpr int kNT  = 64;
constexpr int kPad = kLH + 4; // 260

__global__ __launch_bounds__(256)
void score_kernel(const float* __restrict__ sp, const float* __restrict__ ip,
                  const float* __restrict__ W2, const float* __restrict__ b2,
                  const float* __restrict__ alphas, float* __restrict__ out)
{
    extern __shared__ float smem[];
    float* spt = smem;                       // 16 * 260
    float* ipt = spt + 16 * kPad;            // 64 * 260
    float* w2s = ipt + kNT * kPad;           // 256

    const int tid = threadIdx.x;
    const int n0  = blockIdx.x * kNT;
    const int b0  = blockIdx.y * 16;

    if (tid < kLH) w2s[tid] = alphas[tid >> 6] * W2[tid];
    for (int i = tid; i < 16 * (kLH / 4); i += 256) {      // 1024 float4s
        int r = i >> 6, c4 = i & 63;
        ((float4*)(spt + r * kPad))[c4] =
            ((const float4*)(sp + (size_t)(b0 + r) * kLH))[c4];
    }
    for (int i = tid; i < kNT * (kLH / 4); i += 256) {     // 4096 float4s
        int r = i >> 6, c4 = i & 63;
        int n = n0 + r;
        float4 v = make_float4(0.f, 0.f, 0.f, 0.f);
        if (n < kN) v = ((const float4*)(ip + (size_t)n * kLH))[c4];
        ((float4*)(ipt + r * kPad))[c4] = v;
    }
    __syncthreads();

    float cadd = 0.0f;
    #pragma unroll
    for (int l = 0; l < kL; ++l) cadd += alphas[l] * b2[l];

    const int bl = tid & 15;
    const int nb = (tid >> 4) * 4;
    float a0 = 0.f, a1 = 0.f, a2 = 0.f, a3 = 0.f;
    #pragma unroll 4
    for (int k = 0; k < kLH; ++k) {
        float s = spt[bl * kPad + k];
        float w = w2s[k];
        a0 += fmaxf(s + ipt[(nb + 0) * kPad + k], 0.0f) * w;
        a1 += fmaxf(s + ipt[(nb + 1) * kPad + k], 0.0f) * w;
        a2 += fmaxf(s + ipt[(nb + 2) * kPad + k], 0.0f) * w;
        a3 += fmaxf(s + ipt[(nb + 3) * kPad + k], 0.0f) * w;
    }
    const size_t rowbase = (size_t)(b0 + bl) * kN;
    float acc[4] = {a0, a1, a2, a3};
    #pragma unroll
    for (int j = 0; j < 4; ++j) {
        int n = n0 + nb + j;
        if (n < kN) out[rowbase + n] = acc[j] + cadd;
    }
}

// ============================ loss reduction ==============================
__global__ __launch_bounds__(512)
void loss_kernel(const float* __restrict__ scores,
                 const int* __restrict__ pos_items,
                 const int* __restrict__ neg_items,
                 float* __restrict__ out_loss)
{
    __shared__ float red[512];
    const int b = threadIdx.x;
    const float pos = scores[(size_t)b * kN + pos_items[b]];
    float local = 0.0f;
    #pragma unroll
    for (int j = 0; j < 10; ++j) {
        float neg = scores[(size_t)b * kN + neg_items[b * 10 + j]];
        local += __expf(neg - pos);   // exp(-(pos - neg))
    }
    red[b] = local;
    __syncthreads();
    for (int s = 256; s > 0; s >>= 1) {
        if (b < s) red[b] += red[b + s];
        __syncthreads();
    }
    if (b == 0) out_loss[0] = red[0] / (float)(kB * 10);
}

// ============================== launcher ==================================
extern "C" void kernel_launch(void* const* d_in, const int* in_sizes, int n_in,
                              void* d_out, int out_size, void* d_ws, size_t ws_size,
                              hipStream_t stream)
{
    const int*   user_seq  = (const int*)d_in[1];
    const int*   pos_items = (const int*)d_in[2];
    const int*   neg_items = (const int*)d_in[3];
    const float* emb       = (const float*)d_in[7];
    const float* wih       = (const float*)d_in[8];
    const float* whh       = (const float*)d_in[9];
    const float* bih       = (const float*)d_in[10];
    const float* bhh       = (const float*)d_in[11];
    const float* W1        = (const float*)d_in[12];
    const float* b1        = (const float*)d_in[13];
    const float* W2        = (const float*)d_in[14];
    const float* b2        = (const float*)d_in[15];
    const float* alphas    = (const float*)d_in[16];

    float* scores = (float*)d_out;                 // [B, N]
    float* loss   = scores + (size_t)kB * kN;      // [1]

    float* hbuf = (float*)d_ws;                    // [B, D]     seq_repr
    float* sp   = hbuf + (size_t)kB * kD;          // [B, 256]   (+ b1 folded)
    float* ip   = sp   + (size_t)kB * kLH;         // [N, 256]

    // 1) LSTM: 32 blocks x 16 batch rows, T=50 internal loop
    lstm_kernel<<<kB / 16, 256, 0, stream>>>(user_seq, emb, wih, whh, bih, bhh, hbuf);

    // 2) sp = seq_repr @ W1[:, :D] + b1   (512 tiles / 8 per block)
    wl_gemm_kernel<<<(kB / 16) * (kLH / 16) / 8, 256, 0, stream>>>(
        hbuf, kB, W1, 0, b1, 1, sp);

    // 3) ip = item_emb @ W1[:, D:]        (10000 tiles / 8 per block)
    wl_gemm_kernel<<<(kN / 16) * (kLH / 16) / 8, 256, 0, stream>>>(
        emb, kN, W1, kD, b1, 0, ip);

    // 4) fused relu-dot scores
    dim3 sgrid((kN + kNT - 1) / kNT, kB / 16);
    size_t smem = (size_t)(16 * kPad + kNT * kPad + kLH) * sizeof(float);
    score_kernel<<<sgrid, 256, smem, stream>>>(sp, ip, W2, b2, alphas, scores);

    // 5) loss
    loss_kernel<<<1, 512, 0, stream>>>(scores, pos_items, neg_items, loss);
}